// DynamicEmbeddingBackbone_26414048870692
// MI455X (gfx1250) — compile-verified
//
#include <hip/hip_runtime.h>
#include <hip/hip_bf16.h>

typedef __attribute__((ext_vector_type(2))) float v2f;
typedef __attribute__((ext_vector_type(8))) float v8f;

// Tile config: one block = 256 threads (8 wave32) handles 64 samples.
// Phase 1: gather 8 corner embedding rows per sample, trilinear-weight them,
//          store feat[64][64] to LDS.
// Phase 2: project with W_proj via V_WMMA_F32_16X16X4_F32 (16 k-steps of K=4).
#define ROWS_PER_BLOCK 64
#define LDS_STRIDE 68   // 64 + 4 pad: row stride of 68 floats => bank step of 4

__global__ __launch_bounds__(256)
void dyn_embed_backbone_kernel(const float* __restrict__ values,   // [K,64]
                               const float* __restrict__ p,        // [N,3]
                               const float* __restrict__ Wproj,    // [64,64]
                               const float* __restrict__ bproj,    // [64]
                               const int*   __restrict__ feats,    // [V,8]
                               const int*   __restrict__ svidx,    // [N]
                               float* __restrict__ out,            // [N,64]
                               int N) {
    __shared__ float feat_lds[ROWS_PER_BLOCK * LDS_STRIDE]; // 17408 B
    __shared__ float w_lds[64 * LDS_STRIDE];                // 17408 B

    const int t = threadIdx.x;

    // ---- Stage W_proj (64x64) into LDS: thread t copies 16 floats ----
    {
        const int row = t >> 2;            // 0..63
        const int c0  = (t & 3) << 4;      // 0,16,32,48
        const float4* src = (const float4*)(Wproj + row * 64 + c0);
        float* dst = w_lds + row * LDS_STRIDE + c0;
        #pragma unroll
        for (int j = 0; j < 4; ++j) {
            float4 v = src[j];
            dst[4*j + 0] = v.x; dst[4*j + 1] = v.y;
            dst[4*j + 2] = v.z; dst[4*j + 3] = v.w;
        }
    }

    // ---- Phase 1: gather + trilinear weight into feat_lds ----
    // 4 threads per sample row; each covers 16 of the 64 dims.
    {
        const int r  = t >> 2;             // sample row within block, 0..63
        const int d0 = (t & 3) << 4;       // dim offset 0,16,32,48
        const int n  = blockIdx.x * ROWS_PER_BLOCK + r;

        const int vidx = svidx[n];
        const int* cidx = feats + (size_t)vidx * 8;

        const float px = p[3*n + 0];
        const float py = p[3*n + 1];
        const float pz = p[3*n + 2];
        const float mx = 1.0f - px, my = 1.0f - py, mz = 1.0f - pz;

        // corner order matches _CORNERS in the reference
        float w8[8];
        w8[0] = px * py * pz;  // (1,1,1)
        w8[1] = px * py * mz;  // (1,1,0)
        w8[2] = px * my * pz;  // (1,0,1)
        w8[3] = mx * py * pz;  // (0,1,1)
        w8[4] = px * my * mz;  // (1,0,0)
        w8[5] = mx * py * mz;  // (0,1,0)
        w8[6] = mx * my * pz;  // (0,0,1)
        w8[7] = mx * my * mz;  // (0,0,0)

        float acc[16];
        #pragma unroll
        for (int i = 0; i < 16; ++i) acc[i] = 0.0f;

        #pragma unroll
        for (int c = 0; c < 8; ++c) {
            const float wgt = w8[c];
            const float4* vp = (const float4*)(values + (size_t)cidx[c] * 64 + d0);
            #pragma unroll
            for (int j = 0; j < 4; ++j) {
                float4 v = vp[j];           // global_load_b128 (L2-resident table)
                acc[4*j + 0] = fmaf(wgt, v.x, acc[4*j + 0]);
                acc[4*j + 1] = fmaf(wgt, v.y, acc[4*j + 1]);
                acc[4*j + 2] = fmaf(wgt, v.z, acc[4*j + 2]);
                acc[4*j + 3] = fmaf(wgt, v.w, acc[4*j + 3]);
            }
        }

        float* frow = feat_lds + r * LDS_STRIDE + d0;
        #pragma unroll
        for (int i = 0; i < 16; ++i) frow[i] = acc[i];
    }

    __syncthreads();   // s_barrier_signal/wait on gfx1250

    // ---- Phase 2: out[64x64] = feat @ W via V_WMMA_F32_16X16X4_F32 ----
    // 16 output tiles of 16x16 (4 row-tiles x 4 col-tiles); 8 waves x 2 tiles.
    const int wave = t >> 5;
    const int lane = t & 31;
    const int half = lane >> 4;    // 0: K lanes {0,1} / rows 0-7; 1: K {2,3} / rows 8-15
    const int ln   = lane & 15;

    #pragma unroll
    for (int ti = 0; ti < 2; ++ti) {
        const int tile = wave * 2 + ti;
        const int rt = tile >> 2;        // row tile 0..3
        const int ct = tile & 3;         // col tile 0..3

        v8f c = {};
        const float* arow = feat_lds + (rt * 16 + ln) * LDS_STRIDE; // A: row = M
        const float* bcol = w_lds + ct * 16 + ln;                   // B: col = N

        #pragma unroll
        for (int k = 0; k < 16; ++k) {
            const int kk = 4 * k + 2 * half;
            v2f a, b;
            a[0] = arow[kk];
            a[1] = arow[kk + 1];
            b[0] = bcol[(size_t)kk * LDS_STRIDE];
            b[1] = bcol[(size_t)(kk + 1) * LDS_STRIDE];
            // D = A(16x4) x B(4x16) + C
            c = __builtin_amdgcn_wmma_f32_16x16x4_f32(
                    /*neg_a=*/false, a, /*neg_b=*/false, b,
                    /*c_mod=*/(short)0, c, /*reuse_a=*/false, /*reuse_b=*/false);
        }

        const int col  = ct * 16 + ln;
        const float bias = bproj[col];
        #pragma unroll
        for (int i = 0; i < 8; ++i) {
            const int row = blockIdx.x * ROWS_PER_BLOCK + rt * 16 + half * 8 + i;
            out[(size_t)row * 64 + col] = c[i] + bias;
        }
    }
}

extern "C" void kernel_launch(void* const* d_in, const int* in_sizes, int n_in,
                              void* d_out, int out_size, void* d_ws, size_t ws_size,
                              hipStream_t stream) {
    const float* values = (const float*)d_in[0];   // [K,64]
    const float* p      = (const float*)d_in[1];   // [N,3]
    const float* Wproj  = (const float*)d_in[2];   // [64,64]
    const float* bproj  = (const float*)d_in[3];   // [64]
    const int*   feats  = (const int*)d_in[4];     // [V,8]
    const int*   svidx  = (const int*)d_in[5];     // [N]
    float* out = (float*)d_out;

    const int N = in_sizes[5];                     // 262144
    const int blocks = N / ROWS_PER_BLOCK;         // 4096

    dyn_embed_backbone_kernel<<<blocks, 256, 0, stream>>>(
        values, p, Wproj, bproj, feats, svidx, out, N);
}